// CombinedLoss_73753178407636
// MI455X (gfx1250) — compile-verified
//
#include <hip/hip_runtime.h>
#include <hip/hip_bf16.h>
#include <float.h>

typedef float v2f __attribute__((ext_vector_type(2)));
typedef float v8f __attribute__((ext_vector_type(8)));

#define BATCH 4
#define CHUNK 2048   // reference columns staged in LDS per block iteration (32 KB)

// Pack reference points r -> {-2rx, -2ry, -2rz, |r|^2} so the WMMA B-matrix
// column [K0..K3] comes from a single 128-bit load.
__global__ void pack_points_kernel(const float* __restrict__ pts,
                                   float4* __restrict__ out, int total) {
    int i = blockIdx.x * blockDim.x + threadIdx.x;
    if (i < total) {
        float x = pts[3 * (size_t)i + 0];
        float y = pts[3 * (size_t)i + 1];
        float z = pts[3 * (size_t)i + 2];
        out[i] = make_float4(-2.0f * x, -2.0f * y, -2.0f * z, x * x + y * y + z * z);
    }
}

// Fused pairwise-distance + row-min(+argmin) kernel.
// D = A x B with A row = [qx,qy,qz,1], B col = [-2rx,-2ry,-2rz,|r|^2]
// => D[n][m] = -2 q.r + |r|^2 ; P = D + |q|^2 (added after reduction).
// One wave owns 16 query rows; the block stages reference columns in LDS
// (pre-split per lane-half so the inner loop is a single ds_load_b64).
// A 16x4 f32 layout: lanes0-15 VGPR{0,1}=K{0,1}, lanes16-31 VGPR{0,1}=K{2,3}.
// B 4x16 f32 layout (symmetric): lanes0-15 K{0,1}, lanes16-31 K{2,3}, N=lane%16.
// C/D 16x16 f32: lane -> col=lane%16, VGPR v -> row v + 8*(lane>=16).
template <bool NEED_IDX>
__global__ void chamfer_min_kernel(const float* __restrict__ Q,
                                   const float4* __restrict__ Rp,
                                   float* __restrict__ outMin,
                                   int* __restrict__ outIdx,
                                   int N) {
    __shared__ float2 shB[2 * CHUNK];   // [col][half] -> (K0,K1) or (K2,K3)

    const int lane = threadIdx.x & 31;
    const int wavesPerBlock = blockDim.x >> 5;
    const int waveInBlock = threadIdx.x >> 5;
    const int tiles = N >> 4;
    const int wid = blockIdx.x * wavesPerBlock + waveInBlock;
    const int b  = wid / tiles;          // uniform across the block (8 | tiles)
    const int qt = wid % tiles;
    const int l16 = lane & 15;
    const int hi  = lane >> 4;

    const int qrow = qt * 16 + l16;
    const float* qp = Q + ((size_t)b * N + qrow) * 3;
    const float qx = qp[0], qy = qp[1], qz = qp[2];
    const float qsq = qx * qx + qy * qy + qz * qz;

    v2f a;
    a.x = hi ? qz : qx;     // K2 : K0
    a.y = hi ? 1.0f : qy;   // K3 : K1

    const float4* rb = Rp + (size_t)b * N;

    float minv[8];
    int   mincol[8];
#pragma unroll
    for (int v = 0; v < 8; ++v) { minv[v] = FLT_MAX; mincol[v] = 0; }

    for (int base = 0; base < N; base += CHUNK) {
        // Cooperative stage: split each packed float4 into the two K-pairs.
        for (int i = threadIdx.x; i < CHUNK; i += blockDim.x) {
            float4 r = rb[base + i];
            shB[2 * i + 0] = make_float2(r.x, r.y);   // K0,K1 for lanes 0-15
            shB[2 * i + 1] = make_float2(r.z, r.w);   // K2,K3 for lanes 16-31
        }
        __syncthreads();

        const int ctiles = CHUNK >> 4;
#pragma unroll 2
        for (int t = 0; t < ctiles; ++t) {
            const int lcol = (t << 4) + l16;
            const float2 rv = shB[2 * lcol + hi];     // conflict-free ds_load_b64
            v2f bv;
            bv.x = rv.x;
            bv.y = rv.y;
            v8f c = {0.f, 0.f, 0.f, 0.f, 0.f, 0.f, 0.f, 0.f};
            v8f d = __builtin_amdgcn_wmma_f32_16x16x4_f32(
                false, a, false, bv, (short)0, c, false, false);
            const int col = base + lcol;
#pragma unroll
            for (int v = 0; v < 8; ++v) {
                float dv = d[v];
                if (NEED_IDX) {
                    bool lt = dv < minv[v];
                    mincol[v] = lt ? col : mincol[v];
                    minv[v]   = lt ? dv  : minv[v];
                } else {
                    minv[v] = fminf(minv[v], dv);     // v_min_f32
                }
            }
        }
        __syncthreads();
    }

    // Reduce min(+argmin) over the 16 lanes of each half (xor 1,2,4,8 stays
    // inside the half). Tie-break toward the smaller column (argmin = first).
#pragma unroll
    for (int v = 0; v < 8; ++v) {
        float mv = minv[v];
        int   mc = mincol[v];
#pragma unroll
        for (int s = 1; s < 16; s <<= 1) {
            float ov = __shfl_xor(mv, s, 32);
            int   oc = __shfl_xor(mc, s, 32);
            bool take = (ov < mv) || (NEED_IDX && (ov == mv) && (oc < mc));
            mv = take ? ov : mv;
            mc = take ? oc : mc;
        }
        minv[v] = mv;
        mincol[v] = mc;
    }

    // Lanes 0..7 own rows 0..7 (their own qsq); lanes 24..31 own rows 8..15.
    int v = -1, row = 0;
    if (lane < 8)        { v = lane;      row = qt * 16 + lane; }
    else if (lane >= 24) { v = lane - 24; row = qt * 16 + 8 + (lane - 24); }
    if (v >= 0) {
        outMin[(size_t)b * N + row] = minv[v] + qsq;
        if (NEED_IDX) outIdx[(size_t)b * N + row] = mincol[v];
    }
}

__device__ __forceinline__ float block_reduce_sum(float v, float* sred) {
    const int tid = threadIdx.x;
    sred[tid] = v;
    __syncthreads();
    for (int s = blockDim.x >> 1; s > 0; s >>= 1) {
        if (tid < s) sred[tid] += sred[tid + s];
        __syncthreads();
    }
    float r = sred[0];
    __syncthreads();
    return r;
}

// Single-block deterministic finalize: chamfer means, edge loss, the
// axis-1 (over-edges, per-component) normalization, and cosine loss.
__global__ void finalize_kernel(const float* __restrict__ minsA,
                                const float* __restrict__ minsB,
                                const int* __restrict__ nearest,
                                const float* __restrict__ preds,
                                const float* __restrict__ normals,
                                const int* __restrict__ edges,
                                float* __restrict__ out, int N, int E) {
    __shared__ float sred[1024];
    __shared__ float sInvNN[BATCH][3];
    __shared__ float sInvNV[BATCH][3];
    const int tid = threadIdx.x;
    const int bs  = blockDim.x;

    // chamfer = mean(minsA) + mean(minsB)
    float cs = 0.f;
    for (int i = tid; i < BATCH * N; i += bs) cs += minsA[i] + minsB[i];
    float chamferSum = block_reduce_sum(cs, sred);

    // edge loss + per-(b,d) squared-sum norms over the edge axis
    float edgeAcc = 0.f;
    for (int b = 0; b < BATCH; ++b) {
        float e2x = 0.f, e2y = 0.f, e2z = 0.f;
        float n2x = 0.f, n2y = 0.f, n2z = 0.f;
        for (int e = tid; e < E; e += bs) {
            int e0 = edges[2 * e + 0];
            int e1 = edges[2 * e + 1];
            const float* p0 = preds + ((size_t)b * N + e0) * 3;
            const float* p1 = preds + ((size_t)b * N + e1) * 3;
            float vx = p0[0] - p1[0];
            float vy = p0[1] - p1[1];
            float vz = p0[2] - p1[2];
            edgeAcc += vx * vx + vy * vy + vz * vz;
            e2x += vx * vx; e2y += vy * vy; e2z += vz * vz;
            int ni = nearest[(size_t)b * N + e0];
            const float* nm = normals + ((size_t)b * N + ni) * 3;
            n2x += nm[0] * nm[0]; n2y += nm[1] * nm[1]; n2z += nm[2] * nm[2];
        }
        float sx = block_reduce_sum(e2x, sred);
        float sy = block_reduce_sum(e2y, sred);
        float sz = block_reduce_sum(e2z, sred);
        float tx = block_reduce_sum(n2x, sred);
        float ty = block_reduce_sum(n2y, sred);
        float tz = block_reduce_sum(n2z, sred);
        if (tid == 0) {
            sInvNV[b][0] = 1.0f / fmaxf(sqrtf(sx), 1e-12f);
            sInvNV[b][1] = 1.0f / fmaxf(sqrtf(sy), 1e-12f);
            sInvNV[b][2] = 1.0f / fmaxf(sqrtf(sz), 1e-12f);
            sInvNN[b][0] = 1.0f / fmaxf(sqrtf(tx), 1e-12f);
            sInvNN[b][1] = 1.0f / fmaxf(sqrtf(ty), 1e-12f);
            sInvNN[b][2] = 1.0f / fmaxf(sqrtf(tz), 1e-12f);
        }
    }
    float edgeSum = block_reduce_sum(edgeAcc, sred);
    __syncthreads();

    // cosine loss with the per-component inverse norms
    float cosAcc = 0.f;
    for (int b = 0; b < BATCH; ++b) {
        const float ivx = sInvNV[b][0], ivy = sInvNV[b][1], ivz = sInvNV[b][2];
        const float inx = sInvNN[b][0], iny = sInvNN[b][1], inz = sInvNN[b][2];
        for (int e = tid; e < E; e += bs) {
            int e0 = edges[2 * e + 0];
            int e1 = edges[2 * e + 1];
            const float* p0 = preds + ((size_t)b * N + e0) * 3;
            const float* p1 = preds + ((size_t)b * N + e1) * 3;
            float vx = (p0[0] - p1[0]) * ivx;
            float vy = (p0[1] - p1[1]) * ivy;
            float vz = (p0[2] - p1[2]) * ivz;
            int ni = nearest[(size_t)b * N + e0];
            const float* nm = normals + ((size_t)b * N + ni) * 3;
            float c = nm[0] * inx * vx + nm[1] * iny * vy + nm[2] * inz * vz;
            cosAcc += fabsf(c);
        }
    }
    float cosSum = block_reduce_sum(cosAcc, sred);

    if (tid == 0) {
        float invBN = 1.0f / ((float)BATCH * (float)N);
        float invBE = 1.0f / ((float)BATCH * (float)E);
        float chamfer  = chamferSum * invBN;
        float edgeLoss = edgeSum * invBE;
        float cosLoss  = cosSum * invBE;
        out[0] = 30000.0f * chamfer + 240.0f * edgeLoss + 200000.0f * cosLoss;
    }
}

extern "C" void kernel_launch(void* const* d_in, const int* in_sizes, int n_in,
                              void* d_out, int out_size, void* d_ws, size_t ws_size,
                              hipStream_t stream) {
    const float* preds   = (const float*)d_in[0];  // (B,N,3) f32
    const float* gts     = (const float*)d_in[1];  // (B,N,3) f32
    const float* normals = (const float*)d_in[2];  // (B,N,3) f32
    const int*   edges   = (const int*)d_in[3];    // (E,2)   i32

    const int B = BATCH;
    const int N = in_sizes[0] / (B * 3);   // 8192
    const int E = in_sizes[3] / 2;         // 24576

    // workspace layout
    char* ws = (char*)d_ws;
    float4* Pk    = (float4*)ws;                         // B*N packed preds
    float4* Gk    = Pk + (size_t)B * N;                  // B*N packed gts
    float*  minsA = (float*)(Gk + (size_t)B * N);        // B*N  (mins2)
    float*  minsB = minsA + (size_t)B * N;               // B*N  (mins1)
    int*    nearest = (int*)(minsB + (size_t)B * N);     // B*N

    const int total = B * N;
    pack_points_kernel<<<(total + 255) / 256, 256, 0, stream>>>(preds, Pk, total);
    pack_points_kernel<<<(total + 255) / 256, 256, 0, stream>>>(gts,   Gk, total);

    const int waves = B * (N / 16);        // 2048 wave-tasks per pass
    const int wavesPerBlock = 8;           // 256 threads
    dim3 grid(waves / wavesPerBlock);
    dim3 block(wavesPerBlock * 32);

    // Pass A: Q=gts vs R=preds -> mins2 + nearest_idx
    chamfer_min_kernel<true><<<grid, block, 0, stream>>>(gts, Pk, minsA, nearest, N);
    // Pass B: Q=preds vs R=gts -> mins1 (no argmin)
    chamfer_min_kernel<false><<<grid, block, 0, stream>>>(preds, Gk, minsB, nullptr, N);

    finalize_kernel<<<1, 1024, 0, stream>>>(minsA, minsB, nearest, preds, normals,
                                            edges, (float*)d_out, N, E);
}